// VectorQuantizer_48619029791164
// MI455X (gfx1250) — compile-verified
//
#include <hip/hip_runtime.h>

typedef float v2f __attribute__((ext_vector_type(2)));
typedef float v8f __attribute__((ext_vector_type(8)));

#define VQ_D          256
#define VQ_NROWS      65536
#define VQ_K          4096
#define TILE_CODES    16
#define LDS_STRIDE    260     // 256 + 4 pad floats: spreads ds_load_b64 B-fragment reads across banks
#define KSTEPS        64      // 256 / 4 (K-dim per v_wmma_f32_16x16x4_f32)
#define NTILES        (VQ_K / TILE_CODES)

// ---------------------------------------------------------------------------
// Kernel 1: per-code squared norms ||e_k||^2 ; also zero the loss accumulator.
// One wave per codebook row. grid = 4096/8 blocks of 256.
// ---------------------------------------------------------------------------
__global__ __launch_bounds__(256) void vq_norms(const float* __restrict__ cb,
                                                float* __restrict__ enorm,
                                                float* __restrict__ acc) {
  if (blockIdx.x == 0 && threadIdx.x == 0) *acc = 0.f;
  const int lane = threadIdx.x & 31;
  const int wave = threadIdx.x >> 5;
  const int row  = blockIdx.x * 8 + wave;
  const float4* r4 = (const float4*)(cb + (size_t)row * VQ_D);
  float s = 0.f;
#pragma unroll
  for (int u = 0; u < 2; ++u) {
    float4 v = r4[lane * 2 + u];
    s += v.x * v.x + v.y * v.y + v.z * v.z + v.w * v.w;
  }
#pragma unroll
  for (int m = 16; m >= 1; m >>= 1) s += __shfl_xor(s, m, 32);
  if (lane == 0) enorm[row] = s;
}

// ---------------------------------------------------------------------------
// Kernel 2: WMMA distance GEMM + running argmin.
// Block = 8 waves; wave owns 16 z-rows (A-block resident in 128 VGPRs).
// Loops over 256 code tiles; each tile staged coalesced into padded LDS,
// register-prefetched one tile ahead; 64x v_wmma_f32_16x16x4_f32 per tile.
// ---------------------------------------------------------------------------
__global__ __launch_bounds__(256, 1) void vq_argmin(const float* __restrict__ z,
                                                    const float* __restrict__ cb,
                                                    const float* __restrict__ enorm,
                                                    int* __restrict__ idxout) {
  __shared__ float lds[TILE_CODES * LDS_STRIDE];

  const int tid  = threadIdx.x;
  const int lane = tid & 31;
  const int wave = tid >> 5;
  const int n    = lane & 15;   // column (code) within tile; also A-row selector
  const int h    = lane >> 4;   // lane half -> K offset 2*h in A/B fragments
  const int wave_row = blockIdx.x * 128 + wave * 16;

  // ---- A-block: 16 rows x 256 in WMMA A layout, fully register-resident.
  // lane(l): row M = l&15 ; VGPR pair s holds z[M][4s + 2h], z[M][4s + 2h + 1]
  v2f a[KSTEPS];
  {
    const float* zrow = z + (size_t)(wave_row + n) * VQ_D + 2 * h;
#pragma unroll
    for (int s = 0; s < KSTEPS; ++s)
      a[s] = *(const v2f*)(zrow + 4 * s);
  }

  // Running (min distance, argmin) for the 8 C rows this lane holds:
  // C VGPR j  <->  row M = j + 8*h, column N = lane&15.
  float minv[8];
  int   mini[8];
#pragma unroll
  for (int j = 0; j < 8; ++j) { minv[j] = 3.4e38f; mini[j] = 0; }

  // Prefetch tile 0 into registers (coalesced: 1024 float4 / 256 threads).
  float4 pf[4];
  {
    const float4* src = (const float4*)cb;
#pragma unroll
    for (int i = 0; i < 4; ++i) pf[i] = src[tid + 256 * i];
  }

  for (int t = 0; t < NTILES; ++t) {
    // Stage prefetched tile into padded LDS.
#pragma unroll
    for (int i = 0; i < 4; ++i) {
      const int f = tid + 256 * i;   // float4 index within the 16x256 tile
      const int r = f >> 6;          // code row 0..15  (64 float4 per row)
      const int c = f & 63;          // float4 column
      *(float4*)(&lds[r * LDS_STRIDE + c * 4]) = pf[i];
    }
    __syncthreads();

    // Prefetch next tile while this one computes.
    if (t + 1 < NTILES) {
      const float4* src = (const float4*)(cb + (size_t)(t + 1) * TILE_CODES * VQ_D);
#pragma unroll
      for (int i = 0; i < 4; ++i) pf[i] = src[tid + 256 * i];
    }

    const int code0 = t * TILE_CODES;
    const float en  = enorm[code0 + n];   // ||e||^2 for this lane's code column

    // B fragment for k-step s: lane(l) needs e[n][4s + 2h], e[n][4s + 2h + 1]
    const float* bbase = &lds[n * LDS_STRIDE + 2 * h];

    v8f acc = {};
#pragma unroll
    for (int s = 0; s < KSTEPS; ++s) {
      v2f b = *(const v2f*)(bbase + 4 * s);   // ds_load_b64, bank-spread by pad
      acc = __builtin_amdgcn_wmma_f32_16x16x4_f32(
          /*neg_a=*/false, a[s], /*neg_b=*/false, b,
          /*c_mod=*/(short)0, acc, /*reuse_a=*/false, /*reuse_b=*/false);
    }

    // dist = ||e||^2 - 2 z.e  (||z||^2 constant per row -> irrelevant to argmin)
#pragma unroll
    for (int j = 0; j < 8; ++j) {
      const float dist = en - 2.f * acc[j];
      if (dist < minv[j]) { minv[j] = dist; mini[j] = code0 + n; }
    }
    __syncthreads();   // all waves done reading lds before next tile store
  }

  // Reduce (min, idx) across the 16 lanes of each half (columns of row j+8h),
  // tie-break to the smaller index to match argmin semantics.
#pragma unroll
  for (int j = 0; j < 8; ++j) {
    float v  = minv[j];
    int   ix = mini[j];
#pragma unroll
    for (int m = 1; m <= 8; m <<= 1) {
      const float v2 = __shfl_xor(v, m, 32);
      const int   i2 = __shfl_xor(ix, m, 32);
      if (v2 < v || (v2 == v && i2 < ix)) { v = v2; ix = i2; }
    }
    if (n == 0) idxout[wave_row + j + 8 * h] = ix;
  }
}

// ---------------------------------------------------------------------------
// Kernel 3: gather z_q = codebook[idx] to output + accumulate sum((z_q - z)^2).
// One wave per row; everything is L2-hot after kernel 2.
// ---------------------------------------------------------------------------
__global__ __launch_bounds__(256) void vq_gather(const float* __restrict__ z,
                                                 const float* __restrict__ cb,
                                                 const int* __restrict__ idx,
                                                 float* __restrict__ out,
                                                 float* __restrict__ acc) {
  const int lane = threadIdx.x & 31;
  const int wave = threadIdx.x >> 5;
  const int row  = blockIdx.x * 8 + wave;
  const int k    = idx[row];
  const float4* s4 = (const float4*)(cb + (size_t)k * VQ_D);
  const float4* z4 = (const float4*)(z + (size_t)row * VQ_D);
  float4*       o4 = (float4*)(out + (size_t)row * VQ_D);
  float part = 0.f;
#pragma unroll
  for (int u = 0; u < 2; ++u) {
    const float4 c  = s4[lane * 2 + u];
    const float4 zz = z4[lane * 2 + u];
    o4[lane * 2 + u] = c;
    const float dx = c.x - zz.x, dy = c.y - zz.y, dz = c.z - zz.z, dw = c.w - zz.w;
    part += dx * dx + dy * dy + dz * dz + dw * dw;
  }
#pragma unroll
  for (int m = 16; m >= 1; m >>= 1) part += __shfl_xor(part, m, 32);
  if (lane == 0) atomicAdd(acc, part);
}

// ---------------------------------------------------------------------------
// Kernel 4: scalar loss = BETA*(commit+code) = 0.25 * 2 * mean = 0.5 * sum / (N*d)
// ---------------------------------------------------------------------------
__global__ void vq_finalize(const float* __restrict__ acc,
                            float* __restrict__ out_scalar) {
  *out_scalar = 0.5f * (*acc) / 16777216.f;
}

// ---------------------------------------------------------------------------
extern "C" void kernel_launch(void* const* d_in, const int* in_sizes, int n_in,
                              void* d_out, int out_size, void* d_ws, size_t ws_size,
                              hipStream_t stream) {
  const float* z  = (const float*)d_in[0];   // [16,4096,256] f32
  const float* cb = (const float*)d_in[1];   // [4096,256]   f32
  float* out = (float*)d_out;                // [16*4096*256] z_q_st, then scalar

  float* ws    = (float*)d_ws;
  float* enorm = ws;                         // 4096 floats
  float* acc   = ws + VQ_K;                  // 1 float (padded)
  int*   idxb  = (int*)(ws + VQ_K + 16);     // 65536 ints

  vq_norms  <<<VQ_K / 8,       256, 0, stream>>>(cb, enorm, acc);
  vq_argmin <<<VQ_NROWS / 128, 256, 0, stream>>>(z, cb, enorm, idxb);
  vq_gather <<<VQ_NROWS / 8,   256, 0, stream>>>(z, cb, idxb, out, acc);
  vq_finalize<<<1, 1, 0, stream>>>(acc, out + (size_t)VQ_NROWS * VQ_D);
}